// MKGRec_7473243095346
// MI455X (gfx1250) — compile-verified
//
#include <hip/hip_runtime.h>
#include <hip/hip_bf16.h>
#include <math.h>

// ---------------------------------------------------------------------------
// LightGCN (3 layers) + BPR loss for MI455X (gfx1250, wave32).
// Heavy path: gather -> scale -> scatter-atomic-add SpMM (bandwidth bound,
// ~2.5 GB total -> ~107us @ 23.3 TB/s). Score path uses V_WMMA_F32_16X16X4_F32.
// ---------------------------------------------------------------------------

typedef __attribute__((ext_vector_type(2))) float v2f;
typedef __attribute__((ext_vector_type(8))) float v8f;

#define D_DIM 64

__device__ __forceinline__ void atomAddF(float* p, float v) {
  // native global_atomic_add_f32 (no CAS loop)
  unsafeAtomicAdd(p, v);
}

// ---------------- zero fill (grid-stride) ----------------
__global__ void lgcn_fill0(float* __restrict__ p, size_t n) {
  size_t i = (size_t)blockIdx.x * blockDim.x + threadIdx.x;
  size_t stride = (size_t)gridDim.x * blockDim.x;
  for (; i < n; i += stride) p[i] = 0.0f;
}

// ---------------- in-degree over dst ----------------
__global__ void lgcn_deg(const int* __restrict__ dst, float* __restrict__ deg, int E) {
  int e = blockIdx.x * blockDim.x + threadIdx.x;
  if (e < E) atomAddF(&deg[dst[e]], 1.0f);
}

// ---------------- dinv = deg>0 ? rsqrt(max(deg,1)) : 0  (in place) ----------
__global__ void lgcn_dinv(float* __restrict__ deg, int N) {
  int n = blockIdx.x * blockDim.x + threadIdx.x;
  if (n < N) {
    float d = deg[n];
    deg[n] = (d > 0.0f) ? rsqrtf(fmaxf(d, 1.0f)) : 0.0f;
  }
}

// ---------------- norm[e] = dinv[src]*dinv[dst] ----------------
__global__ void lgcn_norm(const int* __restrict__ src, const int* __restrict__ dst,
                          const float* __restrict__ dinv, float* __restrict__ norm, int E) {
  int e = blockIdx.x * blockDim.x + threadIdx.x;
  if (e < E) norm[e] = dinv[src[e]] * dinv[dst[e]];
}

// ---------------- one LGConv layer: xout[dst] += xin[src]*norm --------------
// half-wave per edge: 16 lanes x float4 = 64 floats, coalesced gather,
// 4 f32 scatter-atomics per lane.
__global__ __launch_bounds__(256)
void lgcn_prop(const int* __restrict__ src, const int* __restrict__ dst,
               const float* __restrict__ norm, const float* __restrict__ xin,
               float* __restrict__ xout, int E) {
  int tid = blockIdx.x * blockDim.x + threadIdx.x;
  int e = tid >> 4;
  if (e >= E) return;
  int q = (tid & 15) << 2;  // float column 0,4,...,60
  int s = src[e];
  int d = dst[e];
  float w = norm[e];
  const float4 v = *(const float4*)(xin + (size_t)s * D_DIM + q);
  float* o = xout + (size_t)d * D_DIM + q;
  atomAddF(o + 0, v.x * w);
  atomAddF(o + 1, v.y * w);
  atomAddF(o + 2, v.z * w);
  atomAddF(o + 3, v.w * w);
}

// ---------------- gather batch rows & accumulate layer outputs --------------
// acc_rows = emb[idx] (init) then += x_layer[idx] after each layer.
__global__ __launch_bounds__(256)
void lgcn_gather_accum(const float* __restrict__ x,
                       const int* __restrict__ ui, const int* __restrict__ pi,
                       const int* __restrict__ ni,
                       float* __restrict__ ua, float* __restrict__ pa,
                       float* __restrict__ na, int B, int init) {
  int tid = blockIdx.x * blockDim.x + threadIdx.x;
  int total = 3 * B * (D_DIM / 4);
  if (tid >= total) return;
  int j = tid >> 4;
  int c = (tid & 15) << 2;
  int idx;
  float* outp;
  if (j < B)            { idx = ui[j];         outp = ua + (size_t)j * D_DIM + c; }
  else if (j < 2 * B)   { idx = pi[j - B];     outp = pa + (size_t)(j - B) * D_DIM + c; }
  else                  { idx = ni[j - 2 * B]; outp = na + (size_t)(j - 2 * B) * D_DIM + c; }
  const float4 v = *(const float4*)(x + (size_t)idx * D_DIM + c);
  if (init) {
    *(float4*)outp = v;
  } else {
    float4 o = *(float4*)outp;
    o.x += v.x; o.y += v.y; o.z += v.z; o.w += v.w;
    *(float4*)outp = o;
  }
}

// ---------------- BPR scores via WMMA f32 16x16x4 ---------------------------
// One wave per 16-sample tile: C_pos = U(16x64) * P^T(64x16), diag = scores.
// A frag layout (16x4 f32): lane l holds A[m=l%16, k=2*(l/16)+r], r=0,1.
// B frag (4x16 f32, = P^T): lane l holds B[k=2*(l/16)+r, n=l%16] = P[n][k],
// i.e. identical addressing into the P row as into the U row.
// Requires B % 64 == 0 (true for B=4096): every wave fully active (EXEC=~0).
__global__ __launch_bounds__(128)
void lgcn_score(const float* __restrict__ ua, const float* __restrict__ pa,
                const float* __restrict__ na, float* __restrict__ cf_acc, int B) {
  __shared__ float lds_pos[4][32][8];
  __shared__ float lds_neg[4][32][8];
  const int lane = threadIdx.x & 31;
  const int wave = threadIdx.x >> 5;
  const int tile = blockIdx.x * 4 + wave;
  const int m = lane & 15;        // sample within tile / column n
  const int h = lane >> 4;        // K-half select
  const size_t row = (size_t)(tile * 16 + m) * D_DIM;
  const float* urow = ua + row;
  const float* prow = pa + row;
  const float* nrow = na + row;

  v8f cpos = {};
  v8f cneg = {};
#pragma unroll
  for (int k0 = 0; k0 < D_DIM; k0 += 4) {
    const int k = k0 + 2 * h;
    v2f a;  a.x  = urow[k]; a.y  = urow[k + 1];
    v2f bp; bp.x = prow[k]; bp.y = prow[k + 1];
    v2f bn; bn.x = nrow[k]; bn.y = nrow[k + 1];
    cpos = __builtin_amdgcn_wmma_f32_16x16x4_f32(false, a, false, bp, (short)0,
                                                 cpos, false, false);
    cneg = __builtin_amdgcn_wmma_f32_16x16x4_f32(false, a, false, bn, (short)0,
                                                 cneg, false, false);
  }
#pragma unroll
  for (int r = 0; r < 8; ++r) {
    lds_pos[wave][lane][r] = cpos[r];
    lds_neg[wave][lane][r] = cneg[r];
  }
  __syncthreads();

  float v = 0.0f;
  if (lane < 16) {
    // diagonal (m,m) of 16x16 f32 C: m<8 -> VGPR m, lane m;
    //                                 m>=8 -> VGPR m-8, lane m+16.
    const int il = (lane < 8) ? lane : lane + 16;
    const int ir = (lane < 8) ? lane : lane - 8;
    const float pd = lds_pos[wave][il][ir];
    const float nd = lds_neg[wave][il][ir];
    // all_layer = acc/4 on both sides -> dot scales by 1/16
    const float s = (nd - pd) * (1.0f / 16.0f);
    v = (s > 0.0f) ? (s + log1pf(expf(-s))) : log1pf(expf(s));  // softplus
  }
#pragma unroll
  for (int off = 16; off > 0; off >>= 1) v += __shfl_xor(v, off, 32);
  if (lane == 0) atomAddF(cf_acc, v);
}

// ---------------- L2 reg on ego embeddings ----------------------------------
__global__ __launch_bounds__(256)
void lgcn_reg(const float* __restrict__ emb, const int* __restrict__ ui,
              const int* __restrict__ pi, const int* __restrict__ ni,
              float* __restrict__ reg_acc, int B) {
  int tid = blockIdx.x * blockDim.x + threadIdx.x;
  int total = 3 * B * (D_DIM / 4);
  float s = 0.0f;
  if (tid < total) {
    int j = tid >> 4;
    int c = (tid & 15) << 2;
    int idx;
    if (j < B)          idx = ui[j];
    else if (j < 2 * B) idx = pi[j - B];
    else                idx = ni[j - 2 * B];
    const float4 v = *(const float4*)(emb + (size_t)idx * D_DIM + c);
    s = v.x * v.x + v.y * v.y + v.z * v.z + v.w * v.w;
  }
#pragma unroll
  for (int off = 16; off > 0; off >>= 1) s += __shfl_xor(s, off, 32);
  if ((threadIdx.x & 31) == 0) atomAddF(reg_acc, s);
}

// ---------------- final scalar ----------------------------------------------
__global__ void lgcn_final(const float* __restrict__ acc2, float* __restrict__ out,
                           float invB) {
  if (threadIdx.x == 0 && blockIdx.x == 0) {
    // CF_WEIGHT=1.0, L2_REG=1e-4
    out[0] = acc2[0] * invB + 1e-4f * 0.5f * acc2[1] * invB;
  }
}

// ---------------------------------------------------------------------------
extern "C" void kernel_launch(void* const* d_in, const int* in_sizes, int n_in,
                              void* d_out, int out_size, void* d_ws, size_t ws_size,
                              hipStream_t stream) {
  const float* emb  = (const float*)d_in[0];
  const int*   edge = (const int*)d_in[1];
  const int*   ui   = (const int*)d_in[2];
  const int*   pi   = (const int*)d_in[3];
  const int*   ni   = (const int*)d_in[4];

  const int N = in_sizes[0] / D_DIM;   // 500,000
  const int E = in_sizes[1] / 2;       // 1,250,000
  const int B = in_sizes[2];           // 4096

  const int* src = edge;
  const int* dst = edge + E;

  // workspace layout (floats)
  float* w    = (float*)d_ws;
  float* deg  = w;                                  // N (becomes dinv in place)
  float* norm = deg + N;                            // E
  float* xa   = norm + E;                           // N*64
  float* xb   = xa + (size_t)N * D_DIM;             // N*64
  float* ua   = xb + (size_t)N * D_DIM;             // B*64
  float* pa   = ua + (size_t)B * D_DIM;             // B*64
  float* na   = pa + (size_t)B * D_DIM;             // B*64
  float* acc2 = na + (size_t)B * D_DIM;             // [cf_sum, reg_sum]

  const int T = 256;
  const size_t ND = (size_t)N * D_DIM;
  const int gatherBlocks = (3 * B * (D_DIM / 4) + T - 1) / T;
  const int propBlocks   = (E * 16 + T - 1) / T;

  // 1. degree + scalar accumulators
  lgcn_fill0<<<1024, T, 0, stream>>>(deg, (size_t)N);
  lgcn_fill0<<<1, 32, 0, stream>>>(acc2, (size_t)2);
  lgcn_deg<<<(E + T - 1) / T, T, 0, stream>>>(dst, deg, E);
  lgcn_dinv<<<(N + T - 1) / T, T, 0, stream>>>(deg, N);
  lgcn_norm<<<(E + T - 1) / T, T, 0, stream>>>(src, dst, deg, norm, E);

  // 2. layer-0 term: acc_rows = emb[idx]
  lgcn_gather_accum<<<gatherBlocks, T, 0, stream>>>(emb, ui, pi, ni, ua, pa, na, B, 1);

  // 3. layer 1: emb -> xa
  lgcn_fill0<<<4096, T, 0, stream>>>(xa, ND);
  lgcn_prop<<<propBlocks, T, 0, stream>>>(src, dst, norm, emb, xa, E);
  lgcn_gather_accum<<<gatherBlocks, T, 0, stream>>>(xa, ui, pi, ni, ua, pa, na, B, 0);

  // 4. layer 2: xa -> xb
  lgcn_fill0<<<4096, T, 0, stream>>>(xb, ND);
  lgcn_prop<<<propBlocks, T, 0, stream>>>(src, dst, norm, xa, xb, E);
  lgcn_gather_accum<<<gatherBlocks, T, 0, stream>>>(xb, ui, pi, ni, ua, pa, na, B, 0);

  // 5. layer 3: xb -> xa
  lgcn_fill0<<<4096, T, 0, stream>>>(xa, ND);
  lgcn_prop<<<propBlocks, T, 0, stream>>>(src, dst, norm, xb, xa, E);
  lgcn_gather_accum<<<gatherBlocks, T, 0, stream>>>(xa, ui, pi, ni, ua, pa, na, B, 0);

  // 6. WMMA tile dot products -> softplus -> cf accumulator
  //    (B=4096 is a multiple of 64 -> every wave fully active, EXEC all ones)
  lgcn_score<<<B / 64, 128, 0, stream>>>(ua, pa, na, &acc2[0], B);

  // 7. L2 reg on ego embeddings
  lgcn_reg<<<gatherBlocks, T, 0, stream>>>(emb, ui, pi, ni, &acc2[1], B);

  // 8. combine scalar
  lgcn_final<<<1, 32, 0, stream>>>(acc2, (float*)d_out, 1.0f / (float)B);
}